// MultiSequenceMultiHeadSelfAttention_72138270703625
// MI455X (gfx1250) — compile-verified
//
#include <hip/hip_runtime.h>
#include <hip/hip_bf16.h>

#define Bc  2
#define Lc  2048
#define HIDc 1024
#define NHc 16
#define HDc 64

typedef __attribute__((ext_vector_type(2))) float v2f;
typedef __attribute__((ext_vector_type(8))) float v8f;
typedef __attribute__((ext_vector_type(4))) unsigned int v4u;
typedef __attribute__((ext_vector_type(4))) int v4i;
typedef __attribute__((ext_vector_type(8))) int v8i;

__device__ __forceinline__ v8f wmma4(v2f a, v2f b, v8f c) {
  // D = A(16x4, f32) x B(4x16, f32) + C(16x16, f32)
  return __builtin_amdgcn_wmma_f32_16x16x4_f32(
      false, a, false, b, (short)0, c, false, false);
}

// ---------------------------------------------------------------------------
// TDM: issue a 2D tile load Global -> LDS via Tensor Data Mover.
// D# built per CDNA5 ISA sec 8: group0 {count=1, lds_addr, global_addr, type=2},
// group1 {data_size=4B, tensor dims, tile dims, dim0 stride}, groups 2/3 zero
// (<=2D tensor). Tracked on TENSORcnt.
// ---------------------------------------------------------------------------
__device__ __forceinline__ void tdm_load_2d(const void* gptr, void* lptr,
                                            unsigned tensor_d0, unsigned tensor_d1,
                                            unsigned tile_d0, unsigned tile_d1,
                                            unsigned long long stride0_elems) {
  unsigned long long ga = (unsigned long long)(size_t)gptr;
  unsigned la = (unsigned)(size_t)lptr;
  v4u g0;
  g0.x = 1u;                                            // count=1 (valid), user mode
  g0.y = la;                                            // lds_addr
  g0.z = (unsigned)ga;                                  // global_addr[31:0]
  g0.w = (unsigned)((ga >> 32) & 0x1FFFFFFu) | (2u << 30); // addr[56:32] | type=2
  v8i g1;
  g1[0] = (int)(2u << 16);                              // data_size=2 (4 bytes)
  g1[1] = (int)((tensor_d0 & 0xFFFFu) << 16);           // tensor_dim0[15:0]
  g1[2] = (int)((tensor_d0 >> 16) | ((tensor_d1 & 0xFFFFu) << 16));
  g1[3] = (int)((tensor_d1 >> 16) | (tile_d0 << 16));   // tile_dim0
  g1[4] = (int)(tile_d1 & 0xFFFFu);                     // tile_dim1, tile_dim2=0
  g1[5] = (int)(unsigned)(stride0_elems & 0xFFFFFFFFu); // tensor_dim0_stride lo
  g1[6] = (int)(unsigned)((stride0_elems >> 32) & 0xFFFFu);
  g1[7] = 0;                                            // tensor_dim1_stride (unused, 2D)
  v4i g2 = {};
  v4i g3 = {};
#if __clang_major__ >= 23
  v8i g4 = {};
  __builtin_amdgcn_tensor_load_to_lds(g0, g1, g2, g3, g4, 0);
#else
  __builtin_amdgcn_tensor_load_to_lds(g0, g1, g2, g3, 0);
#endif
}

// ---------------------------------------------------------------------------
// Kernel 1: positions / segment ids / validity per token (one block per batch)
// ---------------------------------------------------------------------------
__global__ void pos_kernel(const int* __restrict__ aml,
                           int* __restrict__ seg, int* __restrict__ pos,
                           int* __restrict__ valid) {
  __shared__ long long cs[Lc];
  const int b = blockIdx.x;
  if (threadIdx.x == 0) {
    long long acc = 0;
    for (int t = 0; t < Lc; ++t) { acc += (long long)aml[b * Lc + t]; cs[t] = acc; }
  }
  __syncthreads();
  const long long total = cs[Lc - 1];
  for (int t = threadIdx.x; t < Lc; t += blockDim.x) {
    int lo = 0, hi = Lc;
    while (lo < hi) { int mid = (lo + hi) >> 1; if (cs[mid] > (long long)t) hi = mid; else lo = mid + 1; }
    int sg = lo < (Lc - 1) ? lo : (Lc - 1);
    long long start = (sg == 0) ? 0 : cs[sg - 1];
    int vld = ((long long)t < total) ? 1 : 0;
    seg[b * Lc + t]   = sg;
    pos[b * Lc + t]   = vld ? (int)((long long)t - start) : 0;
    valid[b * Lc + t] = vld;
  }
}

// ---------------------------------------------------------------------------
// Kernel 2: QKV GEMM (4096x1024)x(1024x3072)+bias -> Q/K/V [B][NH][L][HD]
// block 256 = 8 waves (4 M x 2 N); block tile 64M x 128N; K chunks of 32
// staged in LDS by the Tensor Data Mover, double buffered.
// ---------------------------------------------------------------------------
__global__ void qkv_gemm(const float* __restrict__ X, const float* __restrict__ W,
                         const float* __restrict__ bias,
                         float* __restrict__ Q, float* __restrict__ K,
                         float* __restrict__ V) {
  __shared__ float xs[2][64][32];    // X tile: 64 rows x 32 k
  __shared__ float wt[2][32][128];   // W tile: 32 k x 128 cols
  const int wave = threadIdx.x >> 5;
  const int lane = threadIdx.x & 31;
  const int half = lane >> 4;
  const int l16  = lane & 15;
  const int wm = wave & 3, wn = wave >> 2;
  const int m0 = blockIdx.x * 64 + wm * 16;
  const int n0base = blockIdx.y * 128 + wn * 64;   // 64-aligned -> one (which,head)

  const float* Xblk = X + (size_t)blockIdx.x * 64 * HIDc;
  const float* Wblk = W + (size_t)blockIdx.y * 128;

  if (wave == 0) {
    tdm_load_2d(Xblk,               &xs[0][0][0], HIDc, 4096, 32, 64, HIDc);
    tdm_load_2d(Wblk,               &wt[0][0][0], 3072, HIDc, 128, 32, 3072);
  }

  v8f acc[4] = {};
  for (int kc = 0; kc < 32; ++kc) {
    const int cur = kc & 1;
    if (wave == 0) {
      if (kc + 1 < 32) {
        tdm_load_2d(Xblk + (size_t)(kc + 1) * 32,        &xs[1 - cur][0][0],
                    HIDc, 4096, 32, 64, HIDc);
        tdm_load_2d(Wblk + (size_t)(kc + 1) * 32 * 3072, &wt[1 - cur][0][0],
                    3072, HIDc, 128, 32, 3072);
        __builtin_amdgcn_s_wait_tensorcnt(2);   // current chunk done
      } else {
        __builtin_amdgcn_s_wait_tensorcnt(0);
      }
    }
    __syncthreads();
#pragma unroll
    for (int kk = 0; kk < 32; kk += 4) {
      const int r = kk + half * 2;
      v2f a = *(const v2f*)&xs[cur][wm * 16 + l16][r];
#pragma unroll
      for (int i = 0; i < 4; ++i) {
        const int c = wn * 64 + i * 16 + l16;
        v2f bf;
        bf.x = wt[cur][r][c];
        bf.y = wt[cur][r + 1][c];
        acc[i] = wmma4(a, bf, acc[i]);
      }
    }
    __syncthreads();   // all waves done reading before buffer is reused
  }

  const int which = n0base >> 10;          // 0=q 1=k 2=v
  const int h = (n0base & 1023) >> 6;
  float* dst = (which == 0) ? Q : ((which == 1) ? K : V);
#pragma unroll
  for (int i = 0; i < 4; ++i) {
    const int col = n0base + i * 16 + l16;
    const int d = col & 63;
    const float bia = bias[col];
#pragma unroll
    for (int r = 0; r < 8; ++r) {
      const int row = m0 + r + half * 8;
      const int b = row >> 11, l = row & (Lc - 1);
      dst[(((b * NHc + h) * Lc) + l) * HDc + d] = acc[i][r] + bia;
    }
  }
}

// ---------------------------------------------------------------------------
// Kernel 3: RoPE on Q and K in place (scale folded into attention)
// ---------------------------------------------------------------------------
__global__ void rope_kernel(float* __restrict__ Q, float* __restrict__ K,
                            const int* __restrict__ pos) {
  int idx = blockIdx.x * blockDim.x + threadIdx.x;
  const int p = idx & 31;        idx >>= 5;
  const int l = idx & (Lc - 1);  idx >>= 11;
  const int h = idx & (NHc - 1); idx >>= 4;
  const int b = idx & 1;
  const int qk = idx >> 1;
  float* base = (qk ? K : Q) + (((b * NHc + h) * Lc + l) * HDc);
  const float po = (float)pos[b * Lc + l];
  const float freq = po * __expf(-(float)(2 * p) * (9.210340371976184f / 64.0f));
  float s, c;
  __sincosf(freq, &s, &c);
  const float x1 = base[p], x2 = base[p + 32];
  base[p]      = x1 * c - x2 * s;
  base[p + 32] = x2 * c + x1 * s;
}

// ---------------------------------------------------------------------------
// Kernel 4: segment-masked flash attention, f32 WMMA, TDM-staged K/V tiles
// grid (L/64, NH, B), block 128 = 4 waves; each wave owns a 16-row Q tile;
// the 4 waves share each 16-key K/V block (4KB each, contiguous) via TDM.
// ---------------------------------------------------------------------------
__global__ void attn_kernel(const float* __restrict__ Q, const float* __restrict__ K,
                            const float* __restrict__ V,
                            const int* __restrict__ seg, const int* __restrict__ valid,
                            float* __restrict__ AO) {
  __shared__ float kt[2][16 * HDc];   // 2 x 4KB
  __shared__ float vt[2][16 * HDc];   // 2 x 4KB
  __shared__ float plds[4][16][17];   // per-wave P tile, padded

  const int wave = threadIdx.x >> 5;
  const int lane = threadIdx.x & 31;
  const int half = lane >> 4;
  const int l16  = lane & 15;
  const int b = blockIdx.z, h = blockIdx.y;
  const int q0 = blockIdx.x * 64 + wave * 16;
  const float* Qb = Q + (size_t)((b * NHc + h) * Lc) * HDc;
  const float* Kb = K + (size_t)((b * NHc + h) * Lc) * HDc;
  const float* Vb = V + (size_t)((b * NHc + h) * Lc) * HDc;
  const int* segb = seg + b * Lc;
  const int* valb = valid + b * Lc;

  // Q tile 16x64 as 16 A-frags (K step 4)
  v2f qf[16];
#pragma unroll
  for (int kb = 0; kb < 16; ++kb)
    qf[kb] = *(const v2f*)(Qb + (q0 + l16) * HDc + kb * 4 + half * 2);

  int segi[8], vali[8];
#pragma unroll
  for (int r = 0; r < 8; ++r) {
    const int row = q0 + r + half * 8;
    segi[r] = segb[row];
    vali[r] = valb[row];
  }

  float m_run[8], l_run[8];
#pragma unroll
  for (int r = 0; r < 8; ++r) { m_run[r] = -3.0e38f; l_run[r] = 0.0f; }
  v8f o[4] = {};

  if (wave == 0) {   // preload block 0 (each 16x64 block is 1024 contiguous floats)
    tdm_load_2d(Kb, &kt[0][0], Lc * HDc, 1, 1024, 1, Lc * HDc);
    tdm_load_2d(Vb, &vt[0][0], Lc * HDc, 1, 1024, 1, Lc * HDc);
  }

  for (int j0 = 0; j0 < Lc; j0 += 16) {
    const int cur = (j0 >> 4) & 1;
    if (wave == 0) {
      if (j0 + 16 < Lc) {
        tdm_load_2d(Kb + (size_t)(j0 + 16) * HDc, &kt[1 - cur][0],
                    Lc * HDc, 1, 1024, 1, Lc * HDc);
        tdm_load_2d(Vb + (size_t)(j0 + 16) * HDc, &vt[1 - cur][0],
                    Lc * HDc, 1, 1024, 1, Lc * HDc);
        __builtin_amdgcn_s_wait_tensorcnt(2);
      } else {
        __builtin_amdgcn_s_wait_tensorcnt(0);
      }
    }
    __syncthreads();

    // ---- S = Q x K^T (16 wmmas over d=64), K frags from LDS ----
    v8f s = {};
#pragma unroll
    for (int kb = 0; kb < 16; ++kb) {
      v2f bf = *(const v2f*)&kt[cur][l16 * HDc + kb * 4 + half * 2];
      s = wmma4(qf[kb], bf, s);
    }
    // ---- mask + scale (1/HD folds both reference scale applications) ----
    const int segj = segb[j0 + l16];
    const int valj = valb[j0 + l16];
    float scol[8];
#pragma unroll
    for (int r = 0; r < 8; ++r) {
      const float v = s[r] * (1.0f / 64.0f);
      const bool ok = (segj == segi[r]) && valj && vali[r];
      scol[r] = ok ? v : -1.0e9f;
    }
    // ---- online softmax; rows live in 16-lane halves ----
    float alpha[8];
#pragma unroll
    for (int r = 0; r < 8; ++r) {
      float v = scol[r];
      for (int off = 1; off < 16; off <<= 1) v = fmaxf(v, __shfl_xor(v, off, 16));
      const float mnew = fmaxf(m_run[r], v);
      const float p = __expf(scol[r] - mnew);
      scol[r] = p;
      float t = p;
      for (int off = 1; off < 16; off <<= 1) t += __shfl_xor(t, off, 16);
      alpha[r] = __expf(m_run[r] - mnew);
      l_run[r] = l_run[r] * alpha[r] + t;
      m_run[r] = mnew;
    }
#pragma unroll
    for (int i = 0; i < 4; ++i)
#pragma unroll
      for (int r = 0; r < 8; ++r) o[i][r] *= alpha[r];

    // ---- P (C-layout) -> LDS -> A-frags; V frags from LDS ----
#pragma unroll
    for (int r = 0; r < 8; ++r) plds[wave][r + half * 8][l16] = scol[r];
    asm volatile("s_wait_dscnt 0" ::: "memory");

#pragma unroll
    for (int kb2 = 0; kb2 < 4; ++kb2) {
      v2f pa = *(const v2f*)&plds[wave][l16][kb2 * 4 + half * 2];
#pragma unroll
      for (int i = 0; i < 4; ++i) {
        const int kk = kb2 * 4 + half * 2;
        v2f vb;
        vb.x = vt[cur][kk * HDc + i * 16 + l16];
        vb.y = vt[cur][(kk + 1) * HDc + i * 16 + l16];
        o[i] = wmma4(pa, vb, o[i]);
      }
    }
    __syncthreads();   // done reading kt/vt[cur] before TDM reuses it
  }

  // ---- normalize + zero invalid rows, write [B][L][NH*HD] ----
#pragma unroll
  for (int i = 0; i < 4; ++i) {
#pragma unroll
    for (int r = 0; r < 8; ++r) {
      const int row = q0 + r + half * 8;
      const float outv = vali[r] ? (o[i][r] / l_run[r]) : 0.0f;
      AO[((size_t)(b * Lc + row) * NHc + h) * HDc + i * 16 + l16] = outv;
    }
  }
}

// ---------------------------------------------------------------------------
// Kernel 5: output projection (4096x1024)x(1024x1024)+bias -> d_out, TDM-staged
// ---------------------------------------------------------------------------
__global__ void out_gemm(const float* __restrict__ X, const float* __restrict__ W,
                         const float* __restrict__ bias, float* __restrict__ Y) {
  __shared__ float xs[2][64][32];
  __shared__ float wt[2][32][128];
  const int wave = threadIdx.x >> 5;
  const int lane = threadIdx.x & 31;
  const int half = lane >> 4;
  const int l16  = lane & 15;
  const int wm = wave & 3, wn = wave >> 2;
  const int m0 = blockIdx.x * 64 + wm * 16;
  const int n0base = blockIdx.y * 128 + wn * 64;

  const float* Xblk = X + (size_t)blockIdx.x * 64 * HIDc;
  const float* Wblk = W + (size_t)blockIdx.y * 128;

  if (wave == 0) {
    tdm_load_2d(Xblk, &xs[0][0][0], HIDc, 4096, 32, 64, HIDc);
    tdm_load_2d(Wblk, &wt[0][0][0], HIDc, HIDc, 128, 32, HIDc);
  }

  v8f acc[4] = {};
  for (int kc = 0; kc < 32; ++kc) {
    const int cur = kc & 1;
    if (wave == 0) {
      if (kc + 1 < 32) {
        tdm_load_2d(Xblk + (size_t)(kc + 1) * 32,        &xs[1 - cur][0][0],
                    HIDc, 4096, 32, 64, HIDc);
        tdm_load_2d(Wblk + (size_t)(kc + 1) * 32 * HIDc, &wt[1 - cur][0][0],
                    HIDc, HIDc, 128, 32, HIDc);
        __builtin_amdgcn_s_wait_tensorcnt(2);
      } else {
        __builtin_amdgcn_s_wait_tensorcnt(0);
      }
    }
    __syncthreads();
#pragma unroll
    for (int kk = 0; kk < 32; kk += 4) {
      const int r = kk + half * 2;
      v2f a = *(const v2f*)&xs[cur][wm * 16 + l16][r];
#pragma unroll
      for (int i = 0; i < 4; ++i) {
        const int c = wn * 64 + i * 16 + l16;
        v2f bf;
        bf.x = wt[cur][r][c];
        bf.y = wt[cur][r + 1][c];
        acc[i] = wmma4(a, bf, acc[i]);
      }
    }
    __syncthreads();
  }

#pragma unroll
  for (int i = 0; i < 4; ++i) {
    const int col = n0base + i * 16 + l16;
    const float bia = bias[col];
#pragma unroll
    for (int r = 0; r < 8; ++r) {
      const int row = m0 + r + half * 8;
      Y[(size_t)row * HIDc + col] = acc[i][r] + bia;
    }
  }
}

// ---------------------------------------------------------------------------
extern "C" void kernel_launch(void* const* d_in, const int* in_sizes, int n_in,
                              void* d_out, int out_size, void* d_ws, size_t ws_size,
                              hipStream_t stream) {
  const float* hidden = (const float*)d_in[0];
  const int*   aml    = (const int*)d_in[1];
  const float* Wqkv   = (const float*)d_in[2];
  const float* bqkv   = (const float*)d_in[3];
  const float* Wout   = (const float*)d_in[4];
  const float* bout   = (const float*)d_in[5];
  float* out = (float*)d_out;

  const size_t QSZ = (size_t)Bc * NHc * Lc * HDc;   // 4,194,304 floats
  float* Qb = (float*)d_ws;
  float* Kb = Qb + QSZ;
  float* Vb = Kb + QSZ;
  float* AO = Vb + QSZ;
  int* seg   = (int*)(AO + QSZ);
  int* pos   = seg + Bc * Lc;
  int* valid = pos + Bc * Lc;

  pos_kernel<<<Bc, 256, 0, stream>>>(aml, seg, pos, valid);
  qkv_gemm<<<dim3(64, 24), 256, 0, stream>>>(hidden, Wqkv, bqkv, Qb, Kb, Vb);
  rope_kernel<<<(2 * Bc * NHc * Lc * 32) / 256, 256, 0, stream>>>(Qb, Kb, pos);
  attn_kernel<<<dim3(Lc / 64, NHc, Bc), 128, 0, stream>>>(Qb, Kb, Vb, seg, valid, AO);
  out_gemm<<<dim3(64, 8), 256, 0, stream>>>(AO, Wout, bout, out);
}